// MipRayMarcher2_45062796870297
// MI455X (gfx1250) — compile-verified
//
#include <hip/hip_runtime.h>
#include <hip/hip_bf16.h>
#include <stdint.h>

// Problem constants (from reference): B=4, R=16384, S=48, C=32
#define BDIM   4
#define RDIM   16384
#define SDIM   48
#define CDIM   32
#define NRAYS  (BDIM * RDIM)        // 65536 rays
#define SM1    (SDIM - 1)           // 47 midpoints
#define WAVE   32
#define WAVES_PER_BLOCK 8
#define BLOCK  (WAVE * WAVES_PER_BLOCK)

// d_out layout (concatenated flat, reference return order):
//   composite_rgb   [B,R,C]    -> NRAYS*CDIM      = 2,097,152 floats
//   composite_depth [B,R,1]    -> NRAYS           =    65,536 floats
//   weights         [B,R,S-1,1]-> NRAYS*SM1       = 3,080,192 floats
#define OUT_RGB_OFF   ((size_t)0)
#define OUT_DEPTH_OFF ((size_t)NRAYS * CDIM)
#define OUT_W_OFF     (OUT_DEPTH_OFF + (size_t)NRAYS)

// ---------------------------------------------------------------------------
// CDNA5 async global->LDS copy (ASYNCcnt path). The clang-22 builtin takes
// (v4i AS1*, v4i AS3*, imm offset, imm cpol) where v4i = int vector_size(16).
// LDS offset = low 32 bits of the generic pointer.
// ---------------------------------------------------------------------------
typedef int v4i __attribute__((vector_size(16)));

__device__ __forceinline__ void async_copy_b128(const float* g, float* l) {
  unsigned loff = (unsigned)(uintptr_t)l;
#if __has_builtin(__builtin_amdgcn_global_load_async_to_lds_b128)
  __builtin_amdgcn_global_load_async_to_lds_b128(
      (__attribute__((address_space(1))) v4i*)(uintptr_t)g,
      (__attribute__((address_space(3))) v4i*)(uintptr_t)loff,
      0, 0);
#else
  asm volatile("global_load_async_to_lds_b128 %0, %1, off"
               :: "v"(loff), "v"((unsigned long long)(uintptr_t)g)
               : "memory");
#endif
}

__device__ __forceinline__ void wait_async0() {
#if __has_builtin(__builtin_amdgcn_s_wait_asynccnt)
  __builtin_amdgcn_s_wait_asynccnt(0);
#else
  asm volatile("s_wait_asynccnt 0" ::: "memory");
#endif
  asm volatile("" ::: "memory");  // compiler fence: keep LDS reads below
}

// ---------------------------------------------------------------------------
// Pass 1a: init global depth min/max accumulators (positive floats -> uint
// bit pattern preserves ordering).
// ---------------------------------------------------------------------------
__global__ void init_minmax_kernel(unsigned* ws) {
  ws[0] = 0x7f800000u;  // +inf  (min accumulator)
  ws[1] = 0x00000000u;  // 0.0f  (max accumulator; depths are in [2,4])
}

// Pass 1b: depths are sorted along S, so global min = min over depths[...,0]
// and global max = max over depths[...,S-1]. Block-reduce then atomics.
__global__ void ray_minmax_kernel(const float* __restrict__ depths,
                                  unsigned* __restrict__ ws) {
  __shared__ unsigned smin[256];
  __shared__ unsigned smax[256];
  const int t = threadIdx.x;
  unsigned mn = 0x7f800000u;
  unsigned mx = 0x00000000u;
  for (int ray = blockIdx.x * blockDim.x + t; ray < NRAYS;
       ray += gridDim.x * blockDim.x) {
    unsigned a = __float_as_uint(depths[(size_t)ray * SDIM]);
    unsigned b = __float_as_uint(depths[(size_t)ray * SDIM + (SDIM - 1)]);
    mn = (a < mn) ? a : mn;
    mx = (b > mx) ? b : mx;
  }
  smin[t] = mn;
  smax[t] = mx;
  __syncthreads();
  for (int off = 128; off > 0; off >>= 1) {
    if (t < off) {
      smin[t] = (smin[t + off] < smin[t]) ? smin[t + off] : smin[t];
      smax[t] = (smax[t + off] > smax[t]) ? smax[t + off] : smax[t];
    }
    __syncthreads();
  }
  if (t == 0) {
    atomicMin(&ws[0], smin[0]);
    atomicMax(&ws[1], smax[0]);
  }
}

// ---------------------------------------------------------------------------
// Main march kernel: one wave32 per ray, lane = color channel.
//   1) issue async global->LDS copy of this ray's 48x32 f32 color tile (6 KB)
//   2) serial weight scan (softplus/alpha/transmittance) overlapped with copy
//   3) s_wait_asynccnt 0, then composite from LDS (bank-conflict-free)
// ---------------------------------------------------------------------------
__global__ __launch_bounds__(BLOCK)
void march_kernel(const float* __restrict__ colors,
                  const float* __restrict__ densities,
                  const float* __restrict__ depths,
                  float* __restrict__ out,
                  const unsigned* __restrict__ ws) {
  __shared__ float lds_col[WAVES_PER_BLOCK][SDIM * CDIM];  // 8 * 6144 B
  __shared__ float lds_w[WAVES_PER_BLOCK][64];             // 8 * 256  B

  const int lane = threadIdx.x & (WAVE - 1);
  const int wv   = threadIdx.x >> 5;
  // Force ray index into an SGPR -> uniform (scalar-path) den/dep addressing.
  const int ray = __builtin_amdgcn_readfirstlane(
      (int)(blockIdx.x * WAVES_PER_BLOCK + wv));

  // ---- 1) async copy: 48*32*4 B = 384 x b128; 32 lanes -> 12 instructions
  const float* gcol = colors + (size_t)ray * (SDIM * CDIM);
  float* lcol = lds_col[wv];
#pragma unroll
  for (int i = 0; i < (SDIM * CDIM) / (WAVE * 4); ++i) {
    const int e = (i * WAVE + lane) * 4;  // float index, 16B granules
    async_copy_b128(gcol + e, lcol + e);
  }

  // ---- 2) transmittance scan (uniform across lanes; hides copy latency)
  const float* den = densities + (size_t)ray * SDIM;
  const float* dep = depths    + (size_t)ray * SDIM;
  float* lw = lds_w[wv];

  float T = 1.0f, dacc = 0.0f, wsum = 0.0f;
  float den_p = den[0];
  float dep_p = dep[0];
  for (int s = 0; s < SM1; ++s) {
    const float den_c = den[s + 1];
    const float dep_c = dep[s + 1];
    const float dm = 0.5f * (den_p + den_c) - 1.0f;
    const float sp = (dm > 20.0f) ? dm : __logf(1.0f + __expf(dm));  // softplus
    const float delta = dep_c - dep_p;
    const float alpha = 1.0f - __expf(-sp * delta);
    const float w = alpha * T;
    T *= (1.0f - alpha + 1e-10f);
    dacc += w * 0.5f * (dep_p + dep_c);
    wsum += w;
    if (lane == 0) lw[s] = w;
    den_p = den_c;
    dep_p = dep_c;
  }
  __builtin_amdgcn_wave_barrier();  // lane0 lw stores ordered before lw reads

  // ---- 3) composite from LDS after async copy lands
  wait_async0();

  // composite_rgb[c] = sum_s col[s][c] * 0.5*(w[s-1]+w[s]),  w[-1]=w[47]=0
  float acc = 0.0f;
  float wprev = 0.0f;
  for (int s = 0; s < SDIM; ++s) {
    const float wcur = (s < SM1) ? lw[s] : 0.0f;
    const float c = lcol[s * CDIM + lane];  // stride-4B across lanes: no conflicts
    acc = __builtin_fmaf(0.5f * (wprev + wcur), c, acc);
    wprev = wcur;
  }
  out[OUT_RGB_OFF + (size_t)ray * CDIM + lane] = acc * 2.0f - 1.0f;

  // weights out: two coalesced b32 stores cover s = 0..46
  out[OUT_W_OFF + (size_t)ray * SM1 + lane] = lw[lane];
  if (lane < SM1 - WAVE)
    out[OUT_W_OFF + (size_t)ray * SM1 + WAVE + lane] = lw[WAVE + lane];

  // composite_depth (uniform): nan -> inf, clip to global [min,max] of depths
  if (lane == 0) {
    const float gmin = __uint_as_float(ws[0]);
    const float gmax = __uint_as_float(ws[1]);
    float cd = dacc / wsum;
    if (cd != cd) cd = __builtin_inff();
    cd = fminf(fmaxf(cd, gmin), gmax);
    out[OUT_DEPTH_OFF + (size_t)ray] = cd;
  }
}

// ---------------------------------------------------------------------------
extern "C" void kernel_launch(void* const* d_in, const int* in_sizes, int n_in,
                              void* d_out, int out_size, void* d_ws, size_t ws_size,
                              hipStream_t stream) {
  (void)in_sizes; (void)n_in; (void)out_size; (void)ws_size;
  const float* colors    = (const float*)d_in[0];
  const float* densities = (const float*)d_in[1];
  const float* depths    = (const float*)d_in[2];
  float* out = (float*)d_out;
  unsigned* ws = (unsigned*)d_ws;

  init_minmax_kernel<<<1, 1, 0, stream>>>(ws);
  ray_minmax_kernel<<<64, 256, 0, stream>>>(depths, ws);
  march_kernel<<<NRAYS / WAVES_PER_BLOCK, BLOCK, 0, stream>>>(
      colors, densities, depths, out, ws);
}